// GlobalSparseAttention_3685081940643
// MI455X (gfx1250) — compile-verified
//
#include <hip/hip_runtime.h>

// ---------------------------------------------------------------------------
// GlobalSparseAttention for MI455X (gfx1250), wave32.
//   - f32 WMMA (V_WMMA_F32_16X16X4_F32) for all GEMMs (exact f32 math keeps
//     the discrete top-k selection faithful to the reference).
//   - Projection GEMMs: LDS double-buffered tiles fed by gfx1250 async
//     global->LDS DMA (global_load_async_to_lds_b128, ASYNCcnt); fragments
//     via ds_load_b64 with bank-conflict-free padding.
//   - Attention scores: Q fragments hoisted to registers, K-tile fragments
//     register double-buffered so global loads overlap the WMMA stream.
//   - Head (b,h) of each projection is the contiguous (2048 x 64) block at
//     float offset (b*16+h)*131072 (the reference's transpose-free reshape).
// ---------------------------------------------------------------------------

typedef float v2f __attribute__((ext_vector_type(2)));
typedef float v8f __attribute__((ext_vector_type(8)));

__device__ __forceinline__ v8f wmma_f32(v2f a, v2f b, v8f c) {
  return __builtin_amdgcn_wmma_f32_16x16x4_f32(
      /*neg_a=*/false, a, /*neg_b=*/false, b,
      /*c_mod=*/(short)0, c, /*reuse_a=*/false, /*reuse_b=*/false);
}

// Async DMA: 16 bytes global -> LDS, tracked by ASYNCcnt (ISA 15.18.3 op 98).
// LDS destination = low 32 bits of the generic pointer.
__device__ __forceinline__ void g2l_async_b128(const float* gptr, float* lptr) {
  const unsigned loff = (unsigned)(unsigned long long)lptr;
  asm volatile("global_load_async_to_lds_b128 %0, %1, off"
               :: "v"(loff), "v"(gptr)
               : "memory");
}

__device__ __forceinline__ void wait_async0() {
#if __has_builtin(__builtin_amdgcn_s_wait_asynccnt)
  __builtin_amdgcn_s_wait_asynccnt(0);
#else
  asm volatile("s_wait_asynccnt 0x0" ::: "memory");
#endif
}

// ---------------------------------------------------------------------------
// Y(MxN) = X(MxK) * W^T (W is NxK row-major) + bias(N).  NT GEMM.
// Block = 256 threads (8 waves). Block tile 128x64, K-tile 32, double
// buffered in LDS.  Wave w computes rows [16w,16w+16) x 64 cols via four
// 16x16 accumulators sharing one A fragment per k-step.
// ---------------------------------------------------------------------------
__global__ __launch_bounds__(256)
void proj_gemm(const float* __restrict__ X, const float* __restrict__ W,
               const float* __restrict__ bias, float* __restrict__ Y,
               int M, int N, int K)
{
  constexpr int BM = 128, BN = 64, BK = 32, PAD = 36;  // PAD: bank-conflict-free
  __shared__ float As[2][BM][PAD];
  __shared__ float Bs[2][BN][PAD];

  const int tid  = threadIdx.x;
  const int lane = tid & 31;
  const int wave = tid >> 5;
  const int lo   = lane & 15;           // row lane within 16x16 tile
  const int hi   = lane >> 4;           // K-pair selector

  const int m0 = blockIdx.x * BM;
  const int n0 = blockIdx.y * BN;

  // Cooperative tile fetch mapping: thread -> (row r, 4-float column c).
  const int lr = tid >> 3;              // 0..31
  const int lc = (tid & 7) * 4;         // 0,4,...,28

  auto issue_tile = [&](int buf, int k0) {
#pragma unroll
    for (int i = 0; i < 4; ++i) {       // A: 128 rows
      const int r = lr + i * 32;
      g2l_async_b128(X + (size_t)(m0 + r) * K + k0 + lc, &As[buf][r][lc]);
    }
#pragma unroll
    for (int i = 0; i < 2; ++i) {       // B(=W rows): 64 rows
      const int r = lr + i * 32;
      g2l_async_b128(W + (size_t)(n0 + r) * K + k0 + lc, &Bs[buf][r][lc]);
    }
  };

  v8f acc0 = {}, acc1 = {}, acc2 = {}, acc3 = {};

  const int NKB = K / BK;
  issue_tile(0, 0);                      // prologue

  for (int kb = 0; kb < NKB; ++kb) {
    const int buf = kb & 1;
    wait_async0();                       // my async writes for tile kb done
    __syncthreads();                     // everyone's writes visible; also
                                         // guards buffer reuse (see below)
    if (kb + 1 < NKB)
      issue_tile(buf ^ 1, (kb + 1) * BK);  // DMA overlaps compute

#pragma unroll
    for (int ks = 0; ks < BK / 4; ++ks) {
      const int kk = ks * 4 + 2 * hi;
      v2f a  = *(const v2f*)&As[buf][wave * 16 + lo][kk];
      v2f w0 = *(const v2f*)&Bs[buf][lo     ][kk];
      v2f w1 = *(const v2f*)&Bs[buf][lo + 16][kk];
      v2f w2 = *(const v2f*)&Bs[buf][lo + 32][kk];
      v2f w3 = *(const v2f*)&Bs[buf][lo + 48][kk];
      acc0 = wmma_f32(a, w0, acc0);
      acc1 = wmma_f32(a, w1, acc1);
      acc2 = wmma_f32(a, w2, acc2);
      acc3 = wmma_f32(a, w3, acc3);
    }
    // No bottom barrier needed: a wave can only issue tile kb+2 into `buf`
    // after the top barrier of iteration kb+1, which every wave reaches only
    // after finishing its reads of `buf` here.
  }

  // C/D layout: VGPR i holds row (i + 8*hi), col lo of the 16x16 tile.
  const int mw = m0 + wave * 16;
  v8f accs[4] = {acc0, acc1, acc2, acc3};
#pragma unroll
  for (int t = 0; t < 4; ++t) {
    const int nc = n0 + t * 16 + lo;
    const float bv = bias[nc];
#pragma unroll
    for (int i = 0; i < 8; ++i)
      Y[(size_t)(mw + i + 8 * hi) * N + nc] = accs[t][i] + bv;
  }
}

// ---------------------------------------------------------------------------
// Per-(b*h) exact top-64 attention.  Block = 256 threads, handles 16 queries.
// Phase 1: WMMA scores (16 x 2048) / 8 into dynamic LDS (128 KB of the WGP's
// 320 KB), Q fragments hoisted, K fragments register double-buffered.
// Phase 2: per query -- 64 x wave-argmax, softmax, weighted V sum.
// ---------------------------------------------------------------------------
__global__ __launch_bounds__(256)
void attn_topk(const float* __restrict__ Qp, const float* __restrict__ Kp,
               const float* __restrict__ Vp, float* __restrict__ Out)
{
  constexpr int SEQ = 2048, D = 64, QT = 16, TK = 64;

  extern __shared__ float lds[];
  float* Ssc = lds;                          // QT x SEQ scores
  float* tv  = lds + QT * SEQ;               // QT x TK top values
  int*   ti  = (int*)(tv + QT * TK);         // QT x TK top indices

  const int lane = threadIdx.x & 31;
  const int wave = threadIdx.x >> 5;
  const int lo   = lane & 15;
  const int hi   = lane >> 4;

  const int bh = blockIdx.y;                 // b*16 + h
  const size_t headOff = (size_t)bh * SEQ * D;
  const float* __restrict__ Qh = Qp + headOff;
  const float* __restrict__ Kh = Kp + headOff;
  const float* __restrict__ Vh = Vp + headOff;
  float* __restrict__ Oh = Out + headOff;

  const int q0 = blockIdx.x * QT;

  // ---- Phase 1: S = (Q_tile . K^T) / sqrt(d) via f32 WMMA ---------------
  // Q fragments for all 16 k-steps, loaded once into registers.
  const float* __restrict__ arow = Qh + (size_t)(q0 + lo) * D;
  v2f qf[16];
#pragma unroll
  for (int ks = 0; ks < 16; ++ks)
    qf[ks] = *(const v2f*)(arow + ks * 4 + 2 * hi);

  auto loadK = [&](v2f (&bf)[16], int n0) {
    const float* __restrict__ brow = Kh + (size_t)(n0 + lo) * D;
#pragma unroll
    for (int ks = 0; ks < 16; ++ks)
      bf[ks] = *(const v2f*)(brow + ks * 4 + 2 * hi);
  };
  auto doTile = [&](const v2f (&bf)[16], int n0) {
    v8f acc = {};
#pragma unroll
    for (int ks = 0; ks < 16; ++ks)
      acc = wmma_f32(qf[ks], bf[ks], acc);
#pragma unroll
    for (int i = 0; i < 8; ++i)
      Ssc[(i + 8 * hi) * SEQ + n0 + lo] = acc[i] * 0.125f;  // 1/sqrt(64)
  };

  // Each wave covers 256 keys = 16 tiles; K fragments double-buffered so the
  // next tile's loads are in flight during the current tile's 16 WMMAs.
  const int base = wave * 256;
  v2f bfA[16], bfB[16];
  loadK(bfA, base);
#pragma unroll 1
  for (int nt = 0; nt < 16; nt += 2) {
    loadK(bfB, base + (nt + 1) * 16);
    doTile(bfA, base + nt * 16);
    if (nt + 2 < 16) loadK(bfA, base + (nt + 2) * 16);
    doTile(bfB, base + (nt + 1) * 16);
  }
  __syncthreads();

  // ---- Phase 2: exact top-64 + softmax + V gather -----------------------
  const float NEG_INF = -__builtin_inff();

  for (int qq = wave; qq < QT; qq += 8) {    // 2 queries per wave
    volatile float* row = Ssc + (size_t)qq * SEQ;
    float* vq = tv + qq * TK;
    int*   iq = ti + qq * TK;

    for (int t = 0; t < TK; ++t) {           // 64 rounds of wave argmax
      float bv = NEG_INF;
      int   bi = lane;
#pragma unroll 4
      for (int j = lane; j < SEQ; j += 32) {
        const float s = row[j];
        if (s > bv) { bv = s; bi = j; }
      }
#pragma unroll
      for (int off = 16; off > 0; off >>= 1) {
        const float ov = __shfl_xor(bv, off, 32);
        const int   oi = __shfl_xor(bi, off, 32);
        if (ov > bv || (ov == bv && oi < bi)) { bv = ov; bi = oi; }
      }
      if (lane == 0) { vq[t] = bv; iq[t] = bi; row[bi] = NEG_INF; }
      __builtin_amdgcn_wave_barrier();
    }

    const float vmax = vq[0];
    const float e0 = __expf(vq[lane]      - vmax);
    const float e1 = __expf(vq[lane + 32] - vmax);
    vq[lane]      = e0;
    vq[lane + 32] = e1;
    float psum = e0 + e1;
#pragma unroll
    for (int off = 16; off > 0; off >>= 1)
      psum += __shfl_xor(psum, off, 32);
    const float inv = 1.0f / psum;
    __builtin_amdgcn_wave_barrier();

    float acc0 = 0.f, acc1 = 0.f;
#pragma unroll 4
    for (int t = 0; t < TK; ++t) {
      const float p = vq[t] * inv;
      const float* __restrict__ vr = Vh + (size_t)iq[t] * D;
      acc0 += p * vr[lane];
      acc1 += p * vr[lane + 32];
    }
    float* __restrict__ orow = Oh + (size_t)(q0 + qq) * D;
    orow[lane]      = acc0;
    orow[lane + 32] = acc1;
  }
}

// ---------------------------------------------------------------------------
// Host-side launch.  Inputs: q,k,v,Wq,bq,Wk,bk,Wv,bv,Wo,bo (f32).
// Workspace: Qp | Kp | Vp | Attn, 16 MB each (64 MB total).
// ---------------------------------------------------------------------------
extern "C" void kernel_launch(void* const* d_in, const int* in_sizes, int n_in,
                              void* d_out, int out_size, void* d_ws, size_t ws_size,
                              hipStream_t stream) {
  const float* q  = (const float*)d_in[0];
  const float* k  = (const float*)d_in[1];
  const float* v  = (const float*)d_in[2];
  const float* Wq = (const float*)d_in[3];
  const float* bq = (const float*)d_in[4];
  const float* Wk = (const float*)d_in[5];
  const float* bk = (const float*)d_in[6];
  const float* Wv = (const float*)d_in[7];
  const float* bv = (const float*)d_in[8];
  const float* Wo = (const float*)d_in[9];
  const float* bo = (const float*)d_in[10];
  float* out = (float*)d_out;

  constexpr int M = 4096, E = 1024;                  // b*s, embed
  constexpr size_t SLAB = (size_t)M * E;             // 4 Mi floats = 16 MB
  float* ws = (float*)d_ws;
  float* Qp = ws;
  float* Kp = ws + SLAB;
  float* Vp = ws + 2 * SLAB;
  float* At = ws + 3 * SLAB;

  const dim3 gGemm(M / 128, E / 64, 1);              // (32, 16)
  const dim3 bGemm(256, 1, 1);

  proj_gemm<<<gGemm, bGemm, 0, stream>>>(q, Wq, bq, Qp, M, E, E);
  proj_gemm<<<gGemm, bGemm, 0, stream>>>(k, Wk, bk, Kp, M, E, E);
  proj_gemm<<<gGemm, bGemm, 0, stream>>>(v, Wv, bv, Vp, M, E, E);

  const size_t ldsBytes = (size_t)16 * 2048 * 4      // scores
                        + (size_t)16 * 64 * 4        // top values
                        + (size_t)16 * 64 * 4;       // top indices
  (void)hipFuncSetAttribute((const void*)attn_topk,
                            hipFuncAttributeMaxDynamicSharedMemorySize,
                            (int)ldsBytes);
  attn_topk<<<dim3(128, 32, 1), dim3(256, 1, 1), ldsBytes, stream>>>(Qp, Kp, Vp, At);

  proj_gemm<<<gGemm, bGemm, 0, stream>>>(At, Wo, bo, out, M, E, E);
}